// MultiBoxLoss_86268713108070
// MI455X (gfx1250) — compile-verified
//
#include <hip/hip_runtime.h>
#include <hip/hip_bf16.h>
#include <math.h>

// MultiBoxLoss (SSD) for MI455X / gfx1250.
// Memory-bound (~190MB traffic, ~8us floor at 23.3 TB/s). CDNA5 features used:
//  * TDM: one tensor_load_to_lds DMAs each 81x128 fp32 plabel tile into LDS
//    (2D descriptor, tensor_dim0 clamp handles the ragged tail), TENSORcnt wait.
//  * Fallback: per-lane global_load_async_to_lds_b128 + s_wait_asynccnt.
// The smooth-L1 VALU work overlaps the DMA; softmax runs out of LDS.

constexpr int kB = 64;
constexpr int kC = 81;
constexpr int kN = 8732;
constexpr int TN  = 128;   // columns per tile in the fused kernel
constexpr int THA = 128;   // threads, fused kernel (4 waves)
constexpr int THB = 1024;  // threads, selection kernel (32 waves)

typedef unsigned int u32x4 __attribute__((ext_vector_type(4)));
typedef int          i32x8 __attribute__((ext_vector_type(8)));
typedef int          i32x4 __attribute__((ext_vector_type(4)));

// ---- CDNA5 async / TDM helpers -------------------------------------------

__device__ __forceinline__ unsigned lds_off_u32(const void* p) {
  // flat addresses of LDS carry the DS byte offset in the low 32 bits
  return (unsigned)(unsigned long long)p;
}

__device__ __forceinline__ void async_load_b128(unsigned lds_byte, const void* g) {
  asm volatile("global_load_async_to_lds_b128 %0, %1, off"
               :: "v"(lds_byte), "v"(g)
               : "memory");
}

__device__ __forceinline__ void async_wait0() {
#if __has_builtin(__builtin_amdgcn_s_wait_asynccnt)
  __builtin_amdgcn_s_wait_asynccnt(0);
#else
  asm volatile("s_wait_asynccnt 0" ::: "memory");
#endif
}

__device__ __forceinline__ void tensor_wait0() {
#if __has_builtin(__builtin_amdgcn_s_wait_tensorcnt)
  __builtin_amdgcn_s_wait_tensorcnt(0);
#else
  asm volatile("s_wait_tensorcnt 0" ::: "memory");
#endif
}

__device__ __forceinline__ float smooth_l1(float x) {
  float ax = fabsf(x);
  return (ax < 1.f) ? 0.5f * ax * ax : ax - 0.5f;
}

// ---- kernel 0: zero per-batch accumulators --------------------------------

__global__ void mbox_zero(float* locloss, float* s1, float* s2, int* numpos) {
  const int t = threadIdx.x;
  if (t < kB) { locloss[t] = 0.f; s1[t] = 0.f; s2[t] = 0.f; numpos[t] = 0; }
}

// ---- kernel A: fused log-softmax CE + smooth-L1 + per-batch partials ------
// grid = (ceil(N/TN), B), block = THA. One thread owns one column n.

__global__ __launch_bounds__(THA) void mbox_fused(
    const float* __restrict__ ploc, const float* __restrict__ plabel,
    const float* __restrict__ gloc, const int* __restrict__ glabel,
    const float* __restrict__ dboxes,
    float* __restrict__ closs, float* __restrict__ locloss,
    float* __restrict__ s1, int* __restrict__ numpos)
{
  __shared__ float sm[kC * TN];   // 81 x 128 tile of plabel (40.5 KB)
  const int b  = blockIdx.y;
  const int n0 = blockIdx.x * TN;
  const int t  = threadIdx.x;
  const int n  = n0 + t;
  const float* pl = plabel + (size_t)b * kC * kN;

#if __has_builtin(__builtin_amdgcn_tensor_load_to_lds)
  // --- TDM path: one DMA for the whole 81 x TN tile -----------------------
  if (t < 32) {  // wave 0 only: TDM ignores EXEC; one issue per workgroup
    const unsigned long long ga = (unsigned long long)(const void*)(pl + n0);
    const unsigned td0 = (unsigned)(kN - n0);  // remaining cols; OOB cols read 0
    u32x4 g0;
    g0.x = 1u;                                          // count=1, user mode
    g0.y = lds_off_u32(sm);                             // lds_addr (bytes)
    g0.z = (unsigned)(ga & 0xFFFFFFFFu);                // global_addr[31:0]
    g0.w = (unsigned)((ga >> 32) & 0x01FFFFFFu)         // global_addr[56:32]
         | (2u << 30);                                  // type = 2 ("image")
    i32x8 g1;
    g1[0] = (2 << 16);                                  // data_size = 4B
    g1[1] = (int)((td0 & 0xFFFFu) << 16);               // tensor_dim0 lo16
    g1[2] = (int)(((td0 >> 16) & 0xFFFFu)               // tensor_dim0 hi16
                  | ((unsigned)kC << 16));              // tensor_dim1 lo16 = 81
    g1[3] = (int)(((unsigned)TN & 0xFFFFu) << 16);      // tile_dim0 = TN
    g1[4] = kC;                                         // tile_dim1=81, tile_dim2=0
    g1[5] = kN;                                         // tensor_dim0_stride = N
    g1[6] = 0;                                          // stride0 hi, stride1 lo
    g1[7] = 0;                                          // stride1 hi
    i32x4 gz4 = {0, 0, 0, 0};                           // groups 2/3: 2D tensor
    i32x8 gz8 = {0, 0, 0, 0, 0, 0, 0, 0};               // extra group (clang-23 form)
    __builtin_amdgcn_tensor_load_to_lds(g0, g1, gz4, gz4, gz8, 0);
  }
#else
  // --- fallback: per-lane async global->LDS copies ------------------------
  const bool full = (n0 + TN <= kN);
  if (full) {
    for (int idx = t; idx < kC * (TN / 4); idx += THA) {
      const int row = idx >> 5;
      const int q   = idx & 31;
      async_load_b128(lds_off_u32(sm + row * TN + q * 4),
                      pl + (size_t)row * kN + n0 + q * 4);
    }
  } else {
    for (int idx = t; idx < kC * TN; idx += THA) {
      const int row = idx >> 7;        // TN == 128
      const int col = idx & (TN - 1);
      const int nn  = n0 + col;
      sm[idx] = (nn < kN) ? pl[(size_t)row * kN + nn] : 0.f;
    }
  }
#endif

  // Overlap: smooth-L1 localization loss while the DMA engine fills LDS.
  float locv = 0.f;
  int   lab  = 0;
  bool  msk  = false;
  if (n < kN) {
    lab = glabel[(size_t)b * kN + n];
    msk = lab > 0;
    const float d0 = dboxes[0 * kN + n], d1 = dboxes[1 * kN + n];
    const float d2 = dboxes[2 * kN + n], d3 = dboxes[3 * kN + n];
    const float g0 = (gloc[((size_t)b * 4 + 0) * kN + n] - d0) / d2;
    const float g1 = (gloc[((size_t)b * 4 + 1) * kN + n] - d1) / d3;
    const float g2 = logf(gloc[((size_t)b * 4 + 2) * kN + n] / d2);
    const float g3 = logf(gloc[((size_t)b * 4 + 3) * kN + n] / d3);
    locv = smooth_l1(ploc[((size_t)b * 4 + 0) * kN + n] - g0)
         + smooth_l1(ploc[((size_t)b * 4 + 1) * kN + n] - g1)
         + smooth_l1(ploc[((size_t)b * 4 + 2) * kN + n] - g2)
         + smooth_l1(ploc[((size_t)b * 4 + 3) * kN + n] - g3);
  }

#if __has_builtin(__builtin_amdgcn_tensor_load_to_lds)
  if (t < 32) tensor_wait0();     // issuing wave drains its TENSORcnt
  __syncthreads();                // publish LDS to all waves
#else
  if (n0 + TN <= kN) async_wait0();
  __syncthreads();
#endif

  // Log-softmax over C=81 from LDS (stride TN -> conflict-free banks),
  // including the target-logit gather (bank = t%64 regardless of lab).
  float cls = 0.f;
  if (n < kN) {
    float mval = -INFINITY;
    for (int c = 0; c < kC; ++c) mval = fmaxf(mval, sm[c * TN + t]);
    float s = 0.f;
    for (int c = 0; c < kC; ++c) s += expf(sm[c * TN + t] - mval);
    const float xt = sm[lab * TN + t];
    cls = mval + logf(s) - xt;               // -log_softmax at target
    closs[(size_t)b * kN + n] = cls;
  }
  __syncthreads();

  // Per-block reduce: loc_loss, S1 = sum(mask*closs), num_pos. Reuse sm.
  sm[t]          = msk ? locv : 0.f;
  sm[TN + t]     = msk ? cls  : 0.f;
  sm[2 * TN + t] = msk ? 1.f  : 0.f;
  __syncthreads();
  for (int off = THA / 2; off > 0; off >>= 1) {
    if (t < off) {
      sm[t]          += sm[t + off];
      sm[TN + t]     += sm[TN + t + off];
      sm[2 * TN + t] += sm[2 * TN + t + off];
    }
    __syncthreads();
  }
  if (t == 0) {
    atomicAdd(&locloss[b], sm[0]);
    atomicAdd(&s1[b], sm[TN]);
    atomicAdd(&numpos[b], (int)(sm[2 * TN] + 0.5f));
  }
}

// ---- kernel B: per-batch hard-negative mining via exact radix-select ------
// grid = B, block = THB. Reproduces stable argsort rank<K semantics:
// pick all con_neg > tau, then ties at tau in ascending index order.

__global__ __launch_bounds__(THB) void mbox_select(
    const float* __restrict__ closs, const int* __restrict__ glabel,
    const int* __restrict__ numpos, float* __restrict__ s2)
{
  __shared__ unsigned key[kN];     // con_neg bit patterns (34.9 KB)
  __shared__ int hist[256];
  __shared__ int sh_digit, sh_rem;
  __shared__ int wsum[32], woff[32];
  __shared__ int chunk_tot, chunk_base;
  __shared__ float red[32];

  const int b = blockIdx.x;
  const int t = threadIdx.x;
  const int lane = t & 31;
  const int wid  = t >> 5;
  const float* cb = closs + (size_t)b * kN;
  const int*   gb = glabel + (size_t)b * kN;

  const int npos = numpos[b];
  const long long k3 = 3LL * npos;
  const int K = (k3 > kN) ? kN : (int)k3;

  // con_neg = mask ? 0 : closs; all values >= 0 so uint bits are order-preserving
  for (int i = t; i < kN; i += THB)
    key[i] = (gb[i] > 0) ? 0u : __float_as_uint(cb[i]);
  __syncthreads();

  float acc = 0.f;
  if (K > 0) {
    // 4-digit MSB radix select of the K-th largest key
    unsigned prefix = 0, pmask = 0;
    int rem = K;
    for (int shift = 24; shift >= 0; shift -= 8) {
      for (int i = t; i < 256; i += THB) hist[i] = 0;
      __syncthreads();
      for (int i = t; i < kN; i += THB) {
        const unsigned k = key[i];
        if ((k & pmask) == prefix) atomicAdd(&hist[(k >> shift) & 255], 1);
      }
      __syncthreads();
      if (t == 0) {
        int cum = 0, d = 255;
        for (;;) {
          const int c = hist[d];
          if (cum + c >= rem || d == 0) break;
          cum += c; --d;
        }
        sh_digit = d; sh_rem = rem - cum;
      }
      __syncthreads();
      prefix |= ((unsigned)sh_digit) << shift;
      pmask  |= 255u << shift;
      rem = sh_rem;
      __syncthreads();
    }
    const unsigned tau = prefix;
    const int need_eq  = rem;   // = K - count(key > tau)

    // ordered scan: take first need_eq elements equal to tau (index order)
    if (t == 0) chunk_base = 0;
    __syncthreads();
    for (int i0 = 0; i0 < kN; i0 += THB) {
      const int i = i0 + t;
      const unsigned k = (i < kN) ? key[i] : 0u;
      const bool gt = (i < kN) && (k > tau);
      const bool eq = (i < kN) && (k == tau);
      if (gt) acc += cb[i];
      const unsigned bm = (unsigned)__ballot(eq);      // wave32 ballot
      const int pin  = __popc(bm & ((1u << lane) - 1u));
      const int wtot = __popc(bm);
      if (lane == 0) wsum[wid] = wtot;
      __syncthreads();
      if (t == 0) {
        int a = 0;
        for (int w = 0; w < 32; ++w) { woff[w] = a; a += wsum[w]; }
        chunk_tot = a;
      }
      __syncthreads();
      const int ordinal = chunk_base + woff[wid] + pin;
      if (eq && ordinal < need_eq) acc += cb[i];
      __syncthreads();
      if (t == 0) chunk_base += chunk_tot;
      __syncthreads();
    }
  }

  // block-reduce acc -> s2[b]
  for (int off = 16; off > 0; off >>= 1) acc += __shfl_down(acc, off, 32);
  if (lane == 0) red[wid] = acc;
  __syncthreads();
  if (wid == 0) {
    float v = red[lane];
    for (int off = 16; off > 0; off >>= 1) v += __shfl_down(v, off, 32);
    if (lane == 0) s2[b] = v;
  }
}

// ---- kernel C: final scalar -----------------------------------------------

__global__ void mbox_final(const float* locloss, const float* s1, const float* s2,
                           const int* numpos, float* out) {
  __shared__ float red[kB];
  const int t = threadIdx.x;
  float v = 0.f;
  if (t < kB) {
    const int np = numpos[t];
    const float tot = locloss[t] + s1[t] + s2[t];
    v = (np > 0) ? tot / fmaxf((float)np, 1e-6f) : 0.f;
  }
  red[t] = v;
  __syncthreads();
  for (int off = kB / 2; off > 0; off >>= 1) {
    if (t < off) red[t] += red[t + off];
    __syncthreads();
  }
  if (t == 0) out[0] = red[0] / (float)kB;
}

// ---- host glue ------------------------------------------------------------

extern "C" void kernel_launch(void* const* d_in, const int* in_sizes, int n_in,
                              void* d_out, int out_size, void* d_ws, size_t ws_size,
                              hipStream_t stream) {
  const float* ploc   = (const float*)d_in[0];
  const float* plabel = (const float*)d_in[1];
  const float* gloc   = (const float*)d_in[2];
  const int*   glabel = (const int*)d_in[3];
  const float* dboxes = (const float*)d_in[4];

  float* ws      = (float*)d_ws;
  float* closs   = ws;                    // B*N floats
  float* locloss = ws + (size_t)kB * kN;  // B floats
  float* s1      = locloss + kB;          // B floats
  float* s2      = s1 + kB;               // B floats
  int*   numpos  = (int*)(s2 + kB);       // B ints

  float* out = (float*)d_out;

  mbox_zero<<<1, kB, 0, stream>>>(locloss, s1, s2, numpos);

  dim3 gridA((kN + TN - 1) / TN, kB);
  mbox_fused<<<gridA, THA, 0, stream>>>(ploc, plabel, gloc, glabel, dboxes,
                                        closs, locloss, s1, numpos);

  mbox_select<<<kB, THB, 0, stream>>>(closs, glabel, numpos, s2);

  mbox_final<<<1, kB, 0, stream>>>(locloss, s1, s2, numpos, out);
}